// Sparsity_60095182405891
// MI455X (gfx1250) — compile-verified
//
#include <hip/hip_runtime.h>
#include <hip/hip_bf16.h>

#ifndef __has_builtin
#define __has_builtin(x) 0
#endif

// x shape: (64, 256, 56, 56) fp32.  HW = 3136, HW/4 = 784 float4 per plane.
// Tile = one float4 of (h,w) for one 8-channel group -> 8 x b128 load/store.
#define TPB      256
#define NT       4                    // tiles per thread (double-buffered pipeline)
#define HW4      784                  // (56*56)/4
#define NTILES   (64 * 32 * HW4)      // 1,605,632
#define NTHREADS (NTILES / NT)        // 401,408
#define NBLOCKS  (NTHREADS / TPB)     // 1,568

typedef int v4i __attribute__((ext_vector_type(4)));
typedef __attribute__((address_space(1))) v4i* g_v4i_ptr;
typedef __attribute__((address_space(3))) v4i* l_v4i_ptr;
typedef __attribute__((address_space(3))) void* las3_ptr;

__device__ __forceinline__ void async_ld_b128(const float4* gsrc, float4* ldst) {
#if __has_builtin(__builtin_amdgcn_global_load_async_to_lds_b128)
  __builtin_amdgcn_global_load_async_to_lds_b128(
      (g_v4i_ptr)(void*)const_cast<float4*>(gsrc), (l_v4i_ptr)(void*)ldst,
      /*offset=*/0, /*cpol=*/0);
#else
  unsigned lds_off = (unsigned)(unsigned long long)(las3_ptr)(void*)ldst;
  asm volatile("global_load_async_to_lds_b128 %0, %1, off"
               :: "v"(lds_off), "v"(gsrc) : "memory");
#endif
}

template <int N>
__device__ __forceinline__ void wait_asynccnt() {
#if __has_builtin(__builtin_amdgcn_s_wait_asynccnt)
  __builtin_amdgcn_s_wait_asynccnt((short)N);
#else
  asm volatile("s_wait_asynccnt %0" :: "i"(N) : "memory");
#endif
}

// 2nd largest of 4 (ties-correct): max(min(hi1,hi2), max(lo1,lo2))
__device__ __forceinline__ float second_of4(float a, float b, float c, float d) {
  float hi1 = fmaxf(a, b), lo1 = fminf(a, b);
  float hi2 = fmaxf(c, d), lo2 = fminf(c, d);
  return fmaxf(fminf(hi1, hi2), fmaxf(lo1, lo2));
}

__device__ __forceinline__ int tile_base_f4(int T) {
  // T = ((n*32 + g) * 784 + hw4); base float4 index = (n*256 + g*8)*784 + hw4
  int hw4  = T % HW4;
  int rest = T / HW4;            // n*32 + g
  return rest * (8 * HW4) + hw4; // max ~12.8M, fits int
}

__global__ __launch_bounds__(TPB) void sparsity_nm_kernel(
    const float* __restrict__ x, float* __restrict__ out) {
  __shared__ float4 smem[2][8][TPB];   // 64 KB
  const int tid = threadIdx.x;
  const int gt  = blockIdx.x * TPB + tid;

  const float4* __restrict__ xv = (const float4*)x;
  float4* __restrict__ ov       = (float4*)out;

  // Prologue: issue tile 0 into buffer 0.
  {
    const int b = tile_base_f4(gt);
#pragma unroll
    for (int c = 0; c < 8; ++c)
      async_ld_b128(xv + b + c * HW4, &smem[0][c][tid]);
  }

#pragma unroll
  for (int t = 0; t < NT; ++t) {
    const int buf = t & 1;
    if (t + 1 < NT) {
      const int bn = tile_base_f4((t + 1) * NTHREADS + gt);
#pragma unroll
      for (int c = 0; c < 8; ++c)
        async_ld_b128(xv + bn + c * HW4, &smem[buf ^ 1][c][tid]);
      wait_asynccnt<8>();   // tile t's 8 copies complete; tile t+1 in flight
    } else {
      wait_asynccnt<0>();
    }
    asm volatile("" ::: "memory");

    float f[8][4];
#pragma unroll
    for (int c = 0; c < 8; ++c) {
      float4 v = smem[buf][c][tid];   // ds_load_b128, conflict-free layout
      f[c][0] = v.x; f[c][1] = v.y; f[c][2] = v.z; f[c][3] = v.w;
    }

    float4 o[8];
#pragma unroll
    for (int j = 0; j < 4; ++j) {
      float a[8];
#pragma unroll
      for (int c = 0; c < 8; ++c) a[c] = fabsf(f[c][j]);
      const float thrA = second_of4(a[0], a[1], a[2], a[3]);
      const float thrB = second_of4(a[4], a[5], a[6], a[7]);
      bool  m1[8];
      float r[8];
#pragma unroll
      for (int c = 0; c < 8; ++c) {
        m1[c] = a[c] >= (c < 4 ? thrA : thrB);
        r[c]  = m1[c] ? 0.0f : a[c];
      }
      float thr2 = r[0];
#pragma unroll
      for (int c = 1; c < 8; ++c) thr2 = fmaxf(thr2, r[c]);
#pragma unroll
      for (int c = 0; c < 8; ++c) {
        const bool keep = m1[c] || (r[c] >= thr2);
        (&o[c].x)[j] = keep ? f[c][j] : 0.0f;
      }
    }

    const int b = tile_base_f4(t * NTHREADS + gt);
#pragma unroll
    for (int c = 0; c < 8; ++c)
      ov[b + c * HW4] = o[c];          // global_store_b128
  }
}

extern "C" void kernel_launch(void* const* d_in, const int* in_sizes, int n_in,
                              void* d_out, int out_size, void* d_ws, size_t ws_size,
                              hipStream_t stream) {
  (void)in_sizes; (void)n_in; (void)out_size; (void)d_ws; (void)ws_size;
  const float* x = (const float*)d_in[0];
  float* out     = (float*)d_out;
  sparsity_nm_kernel<<<NBLOCKS, TPB, 0, stream>>>(x, out);
}